// Attention_76536317215011
// MI455X (gfx1250) — compile-verified
//
#include <hip/hip_runtime.h>
#include <math.h>

typedef __attribute__((ext_vector_type(2))) float v2f;
typedef __attribute__((ext_vector_type(8))) float v8f;

#define SEQ_LEN 8192
#define HIDDEN  4096

// ---------------- Pass 1: energies[s] = dot(enc[s], hidden) --------------
// Uses V_WMMA_F32_16X16X4_F32: A = 16x4 tile of enc rows, B has hidden in
// column 0 only (other columns zeroed), D column 0 accumulates 16 energies.
#define KSPLIT   4
#define KCHUNK   (HIDDEN / KSPLIT)      // 1024 floats per K-slice
#define P1_WAVES 8                      // waves per block
#define P1_ROWS  16                     // rows per wave (WMMA M)

__global__ __launch_bounds__(P1_WAVES * 32)
void energies_wmma_kernel(const float* __restrict__ enc,
                          const float* __restrict__ hidden,
                          float* __restrict__ energies)
{
    __shared__ float sh[KCHUNK];
    const int tid   = threadIdx.x;
    const int kbase = blockIdx.y * KCHUNK;

    // Stage this block's K-slice of hidden into LDS (4 KB).
    for (int i = tid; i < KCHUNK; i += P1_WAVES * 32)
        sh[i] = hidden[kbase + i];
    __syncthreads();

    const int wave  = tid >> 5;
    const int lane  = tid & 31;
    const int m     = lane & 15;          // A-matrix row within the 16-row tile
    const int khalf = (lane >> 4) << 1;   // K offset 0 (lanes 0-15) or 2 (16-31)
    const int row0  = (blockIdx.x * P1_WAVES + wave) * P1_ROWS;

    // A fragment: lane loads contiguous b64 from its row (ISA 16x4 f32 layout).
    const v2f* __restrict__ ap =
        reinterpret_cast<const v2f*>(enc + (size_t)(row0 + m) * HIDDEN + kbase + khalf);
    // Only B column N==0 (lanes 0 and 16 under the mirrored 4x16 layout) is live.
    const float mask = (m == 0) ? 1.0f : 0.0f;

    v8f c = {0.f, 0.f, 0.f, 0.f, 0.f, 0.f, 0.f, 0.f};
    #pragma unroll 4
    for (int k = 0; k < KCHUNK; k += 4) {
        v2f a = ap[k >> 1];                                     // global_load_b64
        v2f b = *reinterpret_cast<const v2f*>(&sh[k + khalf]);  // ds_load_b64
        b.x *= mask;
        b.y *= mask;
        // v_wmma_f32_16x16x4_f32: D = A(16x4) * B(4x16) + C, f32 throughout.
        c = __builtin_amdgcn_wmma_f32_16x16x4_f32(false, a, false, b,
                                                  (short)0, c, false, false);
    }

    // D column N=0: lane 0 holds M=0..7 in c[0..7], lane 16 holds M=8..15.
    if (m == 0) {
        const int rbase = row0 + ((lane >> 4) << 3);
        #pragma unroll
        for (int j = 0; j < 8; ++j)
            atomicAdd(&energies[rbase + j], c[j]);
    }
}

// ---------------- Pass 2: attn = softmax(energies) -----------------------
__global__ __launch_bounds__(1024)
void softmax_kernel(const float* __restrict__ energies,
                    float* __restrict__ attn)
{
    __shared__ float red[32];
    __shared__ float s_bcast;
    const int tid = threadIdx.x;

    float e[8];
    float m = -3.4e38f;
    #pragma unroll
    for (int j = 0; j < 8; ++j) {
        e[j] = energies[tid + 1024 * j];
        m = fmaxf(m, e[j]);
    }
    #pragma unroll
    for (int off = 16; off > 0; off >>= 1) m = fmaxf(m, __shfl_xor(m, off, 32));
    if ((tid & 31) == 0) red[tid >> 5] = m;
    __syncthreads();
    if (tid < 32) {
        float v = red[tid];
        #pragma unroll
        for (int off = 16; off > 0; off >>= 1) v = fmaxf(v, __shfl_xor(v, off, 32));
        if (tid == 0) s_bcast = v;
    }
    __syncthreads();
    m = s_bcast;

    float sum = 0.f;
    #pragma unroll
    for (int j = 0; j < 8; ++j) {
        e[j] = __expf(e[j] - m);
        sum += e[j];
    }
    #pragma unroll
    for (int off = 16; off > 0; off >>= 1) sum += __shfl_xor(sum, off, 32);
    if ((tid & 31) == 0) red[tid >> 5] = sum;
    __syncthreads();
    if (tid < 32) {
        float v = red[tid];
        #pragma unroll
        for (int off = 16; off > 0; off >>= 1) v += __shfl_xor(v, off, 32);
        if (tid == 0) s_bcast = v;
    }
    __syncthreads();

    const float inv = 1.0f / s_bcast;
    #pragma unroll
    for (int j = 0; j < 8; ++j)
        attn[tid + 1024 * j] = e[j] * inv;
}

// ---------------- Pass 3: out[h] = sum_s attn[s] * enc[s][h] -------------
#define P3_CHUNKS 64
#define P3_ROWS   (SEQ_LEN / P3_CHUNKS)  // 128 rows per block

__global__ __launch_bounds__(256)
void weighted_sum_kernel(const float* __restrict__ enc,
                         const float* __restrict__ attn,
                         float* __restrict__ out)
{
    const int tid = threadIdx.x;
    const int s0  = blockIdx.x * P3_ROWS;

    float4 acc[4];
    #pragma unroll
    for (int j = 0; j < 4; ++j) acc[j] = make_float4(0.f, 0.f, 0.f, 0.f);

    for (int s = s0; s < s0 + P3_ROWS; ++s) {
        const float a = attn[s];
        const float4* __restrict__ row =
            reinterpret_cast<const float4*>(enc + (size_t)s * HIDDEN);
        #pragma unroll
        for (int j = 0; j < 4; ++j) {
            float4 v = row[tid + 256 * j];   // global_load_b128, fully coalesced
            acc[j].x = fmaf(a, v.x, acc[j].x);
            acc[j].y = fmaf(a, v.y, acc[j].y);
            acc[j].z = fmaf(a, v.z, acc[j].z);
            acc[j].w = fmaf(a, v.w, acc[j].w);
        }
    }
    #pragma unroll
    for (int j = 0; j < 4; ++j) {
        const int base = tid * 4 + 1024 * j;
        atomicAdd(&out[base + 0], acc[j].x);
        atomicAdd(&out[base + 1], acc[j].y);
        atomicAdd(&out[base + 2], acc[j].z);
        atomicAdd(&out[base + 3], acc[j].w);
    }
}

// ---------------- Launcher ----------------------------------------------
extern "C" void kernel_launch(void* const* d_in, const int* in_sizes, int n_in,
                              void* d_out, int out_size, void* d_ws, size_t ws_size,
                              hipStream_t stream) {
    (void)in_sizes; (void)n_in; (void)out_size; (void)ws_size;

    const float* hidden = (const float*)d_in[0];   // [HIDDEN]
    const float* enc    = (const float*)d_in[1];   // [SEQ_LEN, 1, HIDDEN]
    float*       out    = (float*)d_out;           // [HIDDEN] ++ [SEQ_LEN]
    float*       energ  = (float*)d_ws;            // [SEQ_LEN] scratch

    // Zero atomic-accumulation targets (graph-capture legal).
    hipMemsetAsync(energ, 0, SEQ_LEN * sizeof(float), stream);
    hipMemsetAsync(out,   0, HIDDEN  * sizeof(float), stream);

    dim3 g1(SEQ_LEN / (P1_ROWS * P1_WAVES), KSPLIT);   // (64, 4)
    energies_wmma_kernel<<<g1, P1_WAVES * 32, 0, stream>>>(enc, hidden, energ);

    softmax_kernel<<<1, 1024, 0, stream>>>(energ, out + HIDDEN);

    weighted_sum_kernel<<<P3_CHUNKS, 256, 0, stream>>>(enc, out + HIDDEN, out);
}